// Block_mamba_mini_44925357916652
// MI455X (gfx1250) — compile-verified
//
#include <hip/hip_runtime.h>
#include <math.h>

typedef __attribute__((ext_vector_type(2))) float v2f;
typedef __attribute__((ext_vector_type(8))) float v8f;
typedef __attribute__((ext_vector_type(4))) unsigned int u32x4;
typedef __attribute__((ext_vector_type(8))) int i32x8;
typedef __attribute__((ext_vector_type(4))) int i32x4;

#if defined(__has_builtin)
#if __has_builtin(__builtin_amdgcn_tensor_load_to_lds)
#define USE_TDM 1
#else
#define USE_TDM 0
#endif
#else
#define USE_TDM 0
#endif

#define B_  2
#define L_  4096
#define D_  128
#define ED_ 256
#define N_  16
#define R_  8
#define T_  (B_ * L_)

__device__ __forceinline__ float wave_sum32(float v) {
#pragma unroll
  for (int off = 16; off > 0; off >>= 1) v += __shfl_xor(v, off, 32);
  return v;
}

__device__ __forceinline__ void ln_row(float v[4], int lane,
                                       const float* __restrict__ g,
                                       const float* __restrict__ b) {
  float m = wave_sum32(v[0] + v[1] + v[2] + v[3]) * (1.0f / D_);
  float q = 0.f;
#pragma unroll
  for (int i = 0; i < 4; ++i) { float d = v[i] - m; q += d * d; }
  q = wave_sum32(q) * (1.0f / D_);
  float inv = rsqrtf(q + 1e-5f);
#pragma unroll
  for (int i = 0; i < 4; ++i)
    v[i] = (v[i] - m) * inv * g[lane * 4 + i] + b[lane * 4 + i];
}

// xn = LN_inner(LN_norm1(x)); one wave32 per row of 128
__global__ void ln2_kernel(const float* __restrict__ x,
                           const float* __restrict__ g1, const float* __restrict__ b1,
                           const float* __restrict__ g2, const float* __restrict__ b2,
                           float* __restrict__ out) {
  int row  = blockIdx.x * 4 + (threadIdx.x >> 5);
  int lane = threadIdx.x & 31;
  const float* xr = x + (size_t)row * D_;
  float v[4];
#pragma unroll
  for (int i = 0; i < 4; ++i) v[i] = xr[lane * 4 + i];
  ln_row(v, lane, g1, b1);
  ln_row(v, lane, g2, b2);
#pragma unroll
  for (int i = 0; i < 4; ++i) out[(size_t)row * D_ + lane * 4 + i] = v[i];
}

// x1 = x + attn (stored); h1 = LN_norm1(x1)
__global__ void add_ln_kernel(const float* __restrict__ x, const float* __restrict__ attn,
                              const float* __restrict__ g, const float* __restrict__ b,
                              float* __restrict__ x1, float* __restrict__ h1) {
  int row  = blockIdx.x * 4 + (threadIdx.x >> 5);
  int lane = threadIdx.x & 31;
  const float* xr = x    + (size_t)row * D_;
  const float* ar = attn + (size_t)row * D_;
  float v[4];
#pragma unroll
  for (int i = 0; i < 4; ++i) {
    v[i] = xr[lane * 4 + i] + ar[lane * 4 + i];
    x1[(size_t)row * D_ + lane * 4 + i] = v[i];
  }
  ln_row(v, lane, g, b);
#pragma unroll
  for (int i = 0; i < 4; ++i) h1[(size_t)row * D_ + lane * 4 + i] = v[i];
}

// ---------------------------------------------------------------------------
// in_proj GEMM with TDM-staged A tile:
//   grid: (T/16) m-tiles x 4 n-blocks; 256 threads = 8 waves, each wave one
//   16x16 tile of C. A tile (16x128 fp32, 8KB) DMA'd into LDS by the Tensor
//   Data Mover with 2-dword row padding (row stride 130 floats -> no bank
//   conflicts), shared by all 8 waves (8x A reuse).
// ---------------------------------------------------------------------------
__global__ void __launch_bounds__(256)
gemm_inproj_tdm(const float* __restrict__ A, const float* __restrict__ W,
                float* __restrict__ C) {
  __shared__ float sA[16 * 130];  // padded rows

  int tm = blockIdx.x >> 2;
  int tn = ((blockIdx.x & 3) << 3) + (threadIdx.x >> 5);
  int lane = threadIdx.x & 31;
  int lo = lane & 15, hi = lane >> 4;

  if (threadIdx.x < 32) {
#if USE_TDM
    unsigned long long ga =
        (unsigned long long)(const void*)(A + (size_t)tm * 16 * D_);
    // LDS byte offset of sA: low 32 bits of its flat address (LDS aperture).
    unsigned int lds_off = (unsigned int)(unsigned long long)(void*)&sA[0];
    // D# group 0: count=1 | lds_addr | global_addr(57b) | type=2
    u32x4 g0;
    g0[0] = 1u;                                   // count=1, user mode, no gather
    g0[1] = lds_off;                              // lds_addr = base of sA
    g0[2] = (unsigned int)(ga & 0xFFFFFFFFu);     // global_addr[31:0]
    g0[3] = (unsigned int)((ga >> 32) & 0x01FFFFFFu) | (2u << 30);  // [56:32], type=2
    // D# group 1: data_size=4B, pad 2 dwords per 128 dwords,
    //             tensor 128 x 8192, tile 128 x 16, stride0 = 128
    i32x8 g1;
    g1[0] = (int)((2u << 16) | (1u << 20) | (6u << 22) | (1u << 25));
    g1[1] = (int)((128u & 0xFFFFu) << 16);        // tensor_dim0 lo16
    g1[2] = (int)(((unsigned)T_ & 0xFFFFu) << 16);// tensor_dim0 hi=0 | tensor_dim1 lo16
    g1[3] = (int)((128u & 0xFFFFu) << 16);        // tensor_dim1 hi=0 | tile_dim0=128
    g1[4] = (int)16u;                             // tile_dim1=16, tile_dim2=0
    g1[5] = (int)128u;                            // tensor_dim0_stride lo32
    g1[6] = 0;                                    // stride0 hi | stride1 lo
    g1[7] = 0;
    i32x4 gz = {0, 0, 0, 0};
    i32x8 gz8 = {0, 0, 0, 0, 0, 0, 0, 0};
    __builtin_amdgcn_tensor_load_to_lds(g0, g1, gz, gz, gz8, 0);
    __builtin_amdgcn_s_wait_tensorcnt(0);
    // TDM wrote sA behind the compiler's back -- forbid folding sA loads.
    asm volatile("" ::: "memory");
#else
    for (int i = lane; i < 16 * 32; i += 32) {    // fallback: cooperative copy
      int r = i >> 5, c4 = (i & 31) * 4;
      const float* src = A + ((size_t)tm * 16 + r) * D_ + c4;
#pragma unroll
      for (int j = 0; j < 4; ++j) sA[r * 130 + c4 + j] = src[j];
    }
#endif
  }
  __syncthreads();
  asm volatile("" ::: "memory");  // sA contents produced by DMA, not visible stores

  int ncol = tn * 16 + lo;
  const float* Wr = W + (size_t)ncol * D_ + 2 * hi;
  const float* Ar = sA + lo * 130 + 2 * hi;

  v8f acc = {0.f, 0.f, 0.f, 0.f, 0.f, 0.f, 0.f, 0.f};
#pragma unroll 4
  for (int k = 0; k < D_; k += 4) {
    v2f a, b;
    a.x = Ar[k];
    a.y = Ar[k + 1];
    b.x = Wr[k];
    b.y = Wr[k + 1];
    acc = __builtin_amdgcn_wmma_f32_16x16x4_f32(false, a, false, b, (short)0, acc,
                                                false, false);
  }
#pragma unroll
  for (int v = 0; v < 8; ++v) {
    int m = tm * 16 + v + 8 * hi;
    C[(size_t)m * 512 + ncol] = acc[v];
  }
}

// Generic WMMA fp32 GEMM: C[m,n] = epilogue( sum_k A[m,k]*W[n,k] )
// EPI: 0=plain 1=softplus(+bias) 2=gelu_exact(+bias) 3=bias+residual
template <int EPI, bool NG>
__global__ void gemm_wmma(const float* __restrict__ A, const float* __restrict__ W,
                          float* __restrict__ C,
                          const float* __restrict__ bias, const float* __restrict__ resid,
                          int tilesN, int Kdim, int lda, int ldc, int Nout) {
  int tile = blockIdx.x * 4 + (threadIdx.x >> 5);
  int tm = tile / tilesN, tn = tile % tilesN;
  int lane = threadIdx.x & 31;
  int lo = lane & 15, hi = lane >> 4;

  const float* Ar = A + (size_t)(tm * 16 + lo) * lda + 2 * hi;
  int ncol = tn * 16 + lo;
  bool nok = (!NG) || (ncol < Nout);
  const float* Wr = W + (size_t)(nok ? ncol : 0) * Kdim + 2 * hi;

  v8f acc = {0.f, 0.f, 0.f, 0.f, 0.f, 0.f, 0.f, 0.f};
  for (int k = 0; k < Kdim; k += 4) {
    v2f a, b;
    a.x = Ar[k];
    a.y = Ar[k + 1];
    float w0 = Wr[k], w1 = Wr[k + 1];
    b.x = nok ? w0 : 0.f;
    b.y = nok ? w1 : 0.f;
    acc = __builtin_amdgcn_wmma_f32_16x16x4_f32(false, a, false, b, (short)0, acc,
                                                false, false);
  }
  if (NG && !nok) return;
#pragma unroll
  for (int v = 0; v < 8; ++v) {
    int m = tm * 16 + v + 8 * hi;
    float val = acc[v];
    if (EPI == 1) {  // softplus(x + bias)
      val += bias[ncol];
      val = (val > 20.f) ? val : log1pf(__expf(val));
    } else if (EPI == 2) {  // exact GELU
      val += bias[ncol];
      val = 0.5f * val * (1.0f + erff(val * 0.70710678118654752f));
    } else if (EPI == 3) {  // bias + residual
      val += bias[ncol] + resid[(size_t)m * ldc + ncol];
    }
    C[(size_t)m * ldc + ncol] = val;
  }
}

// causal depthwise conv (K=4) + bias + SiLU; u half of xz -> u_act
__global__ void conv_silu_kernel(const float* __restrict__ xz,
                                 const float* __restrict__ cw,
                                 const float* __restrict__ cb,
                                 float* __restrict__ u) {
  int idx = blockIdx.x * blockDim.x + threadIdx.x;  // T_*ED_
  int e = idx & (ED_ - 1);
  int t = idx >> 8;
  int l = t & (L_ - 1);
  float acc = cb[e];
#pragma unroll
  for (int k = 0; k < 4; ++k) {
    int ls = l - 3 + k;
    if (ls >= 0) acc = fmaf(cw[e * 4 + k], xz[(size_t)(t - 3 + k) * 512 + e], acc);
  }
  float sil = acc / (1.0f + __expf(-acc));
  u[(size_t)t * ED_ + e] = sil;
}

// selective scan: lanes = (e pair, n); 16-lane reduction for y = sum_n h*C
__global__ void scan_kernel(const float* __restrict__ delta, const float* __restrict__ dbc,
                            const float* __restrict__ u, const float* __restrict__ xz,
                            const float* __restrict__ A_log, const float* __restrict__ Dp,
                            float* __restrict__ y) {
  int w = (blockIdx.x * blockDim.x + threadIdx.x) >> 5;  // 256 waves total
  int lane = threadIdx.x & 31;
  int lo = lane & 15, hi = lane >> 4;
  int b = w >> 7;
  int e = ((w & 127) << 1) + hi;
  float Aen = -__expf(A_log[e * N_ + lo]);
  float Dpe = Dp[e];
  float h = 0.0f;
  int tbase = b * L_;
  for (int l = 0; l < L_; ++l) {
    int t = tbase + l;
    const float* dbct = dbc + (size_t)t * 40;
    if ((l & 15) == 0)
      __builtin_prefetch((const void*)(dbct + 40 * 16), 0, 1);  // global_prefetch
    float dlt = delta[(size_t)t * ED_ + e];
    float uv  = u[(size_t)t * ED_ + e];
    float Bn  = dbct[8 + lo];
    float Cn  = dbct[24 + lo];
    float dA  = __expf(dlt * Aen);
    h = fmaf(dA, h, dlt * Bn * uv);
    float p = h * Cn;
    p += __shfl_xor(p, 1, 32);
    p += __shfl_xor(p, 2, 32);
    p += __shfl_xor(p, 4, 32);
    p += __shfl_xor(p, 8, 32);
    if (lo == 0) {
      float zv = xz[(size_t)t * 512 + 256 + e];
      float sil = zv / (1.0f + __expf(-zv));
      y[(size_t)t * ED_ + e] = (p + uv * Dpe) * sil;
    }
  }
}

extern "C" void kernel_launch(void* const* d_in, const int* in_sizes, int n_in,
                              void* d_out, int out_size, void* d_ws, size_t ws_size,
                              hipStream_t stream) {
  (void)in_sizes; (void)n_in; (void)out_size; (void)ws_size;
  const float* x       = (const float*)d_in[0];
  const float* norm1_g = (const float*)d_in[1];
  const float* norm1_b = (const float*)d_in[2];
  const float* inner_g = (const float*)d_in[3];
  const float* inner_b = (const float*)d_in[4];
  const float* in_w    = (const float*)d_in[5];   // [512,128]
  const float* conv_w  = (const float*)d_in[6];   // [256,1,4]
  const float* conv_b  = (const float*)d_in[7];
  const float* xproj_w = (const float*)d_in[8];   // [40,256]
  const float* dt_w    = (const float*)d_in[9];   // [256,8]
  const float* dt_b    = (const float*)d_in[10];
  const float* A_log   = (const float*)d_in[11];  // [256,16]
  const float* Dp      = (const float*)d_in[12];
  const float* out_w   = (const float*)d_in[13];  // [128,256]
  const float* fc1_w   = (const float*)d_in[14];  // [256,128]
  const float* fc1_b   = (const float*)d_in[15];
  const float* fc2_w   = (const float*)d_in[16];  // [128,256]
  const float* fc2_b   = (const float*)d_in[17];
  float* out = (float*)d_out;

  float* ws    = (float*)d_ws;
  float* xn    = ws;                         // T*128
  float* xz    = xn    + (size_t)T_ * 128;   // T*512
  float* ubuf  = xz    + (size_t)T_ * 512;   // T*256
  float* dbc   = ubuf  + (size_t)T_ * 256;   // T*40
  float* dlt   = dbc   + (size_t)T_ * 40;    // T*256
  float* ybuf  = dlt   + (size_t)T_ * 256;   // T*256
  float* attn  = ybuf  + (size_t)T_ * 256;   // T*128
  float* x1    = attn  + (size_t)T_ * 128;   // T*128
  float* h1    = x1    + (size_t)T_ * 128;   // T*128
  float* gbuf  = h1    + (size_t)T_ * 128;   // T*256

  // 1. xn = LN_inner(LN_norm1(x))
  ln2_kernel<<<T_ / 4, 128, 0, stream>>>(x, norm1_g, norm1_b, inner_g, inner_b, xn);
  // 2. in_proj via TDM-staged A: [T,128] x [512,128]^T -> xz [T,512]
  gemm_inproj_tdm<<<(T_ / 16) * 4, 256, 0, stream>>>(xn, in_w, xz);
  // 3. depthwise causal conv + SiLU on u half
  conv_silu_kernel<<<(T_ * ED_) / 256, 256, 0, stream>>>(xz, conv_w, conv_b, ubuf);
  // 4. x_proj: [T,256] x [40,256]^T -> dbc [T,40]  (N padded to 48, guarded)
  gemm_wmma<0, true><<<(T_ / 16) * 3 / 4, 128, 0, stream>>>(
      ubuf, xproj_w, dbc, nullptr, nullptr, 3, 256, 256, 40, 40);
  // 5. delta = softplus(dt @ dt_w^T + dt_b): A = dbc[:, :8] (lda=40), K=8
  gemm_wmma<1, false><<<(T_ / 16) * 16 / 4, 128, 0, stream>>>(
      dbc, dt_w, dlt, dt_b, nullptr, 16, 8, 40, 256, 256);
  // 6. selective scan (+ skip D, * silu(z)) -> ybuf [T,256]
  scan_kernel<<<32, 256, 0, stream>>>(dlt, dbc, ubuf, xz, A_log, Dp, ybuf);
  // 7. out_proj: [T,256] x [128,256]^T -> attn [T,128]
  gemm_wmma<0, false><<<(T_ / 16) * 8 / 4, 128, 0, stream>>>(
      ybuf, out_w, attn, nullptr, nullptr, 8, 256, 256, 128, 128);
  // 8. x1 = x + attn ; h1 = LN_norm1(x1)
  add_ln_kernel<<<T_ / 4, 128, 0, stream>>>(x, attn, norm1_g, norm1_b, x1, h1);
  // 9. fc1 + exact GELU -> gbuf [T,256]
  gemm_wmma<2, false><<<(T_ / 16) * 16 / 4, 128, 0, stream>>>(
      h1, fc1_w, gbuf, fc1_b, nullptr, 16, 128, 128, 256, 256);
  // 10. fc2 + bias + residual(x1) -> out [T,128]
  gemm_wmma<3, false><<<(T_ / 16) * 8 / 4, 128, 0, stream>>>(
      gbuf, fc2_w, out, fc2_b, x1, 8, 256, 256, 128, 128);
}